// depthwise_separable_conv_26568667693196
// MI455X (gfx1250) — compile-verified
//
#include <hip/hip_runtime.h>
#include <cstdint>

// AdderNet L1-distance conv pair for MI455X (gfx1250, wave32).
//   Kernel A: 3x3 pad1, 128->128, N=8, 64x64  (x, w_dw) -> mid (d_ws)
//   Kernel B: 1x1,      128->256              (mid, w_pw) -> out
// Both are f32 VALU-bound (acc -= |a-b| == v_sub + v_sub with |src| modifier).
// Kernel B stages weights via the Tensor Data Mover (tensor_load_to_lds with
// row padding) and activations via global_load_async_to_lds_b128.

typedef unsigned int v4u __attribute__((ext_vector_type(4)));
typedef int v4i __attribute__((ext_vector_type(4)));
typedef int v8i __attribute__((ext_vector_type(8)));

// ---------------------------------------------------------------- Kernel A --
// Tile: 64 co x 64 pixels (8x8 spatial). 256 threads, each 4co x 4pix accums.
// LDS: full-depth x halo tile [128ci][10r][10c] + transposed weight chunk
//      [16ci*9][64co] (float4 reads, conflict-free).
__global__ __launch_bounds__(256) void adder3x3_kernel(
    const float* __restrict__ x, const float* __restrict__ w,
    float* __restrict__ out)
{
  __shared__ __align__(16) float xs[128 * 100];   // 51200 B
  __shared__ __align__(16) float wsT[144 * 64];   // 36864 B

  const int t  = threadIdx.x;
  const int cg = t & 15;          // co group   (4 consecutive co)
  const int pg = t >> 4;          // pixel group(4 consecutive cols)
  const int pr = pg >> 1;         // pixel row 0..7
  const int pc = (pg & 1) * 4;    // pixel col base 0 or 4

  int b = blockIdx.x;             // [2 coT][8 n][64 tiles]
  const int coT = b >> 9;  b &= 511;
  const int n   = b >> 6;
  const int tl  = b & 63;
  const int h0  = (tl >> 3) * 8;
  const int w0  = (tl & 7) * 8;
  const int co0 = coT * 64;

  const float* xn = x + (size_t)n * 128 * 4096;

  // Stage x halo tile (zero-filled borders), all 128 channels.
  for (int i = t; i < 12800; i += 256) {
    int ci  = i / 100;
    int rem = i - ci * 100;
    int r   = rem / 10;
    int c   = rem - r * 10;
    int h   = h0 + r - 1;
    int ww  = w0 + c - 1;
    float v = 0.0f;
    if (h >= 0 && h < 64 && ww >= 0 && ww < 64)
      v = xn[ci * 4096 + h * 64 + ww];
    xs[i] = v;
  }

  float acc[4][4];
  #pragma unroll
  for (int j = 0; j < 4; ++j)
    #pragma unroll
    for (int i = 0; i < 4; ++i) acc[j][i] = 0.0f;

  for (int cc = 0; cc < 128; cc += 16) {
    __syncthreads();  // xs ready (first iter) / wsT reads of prev chunk done
    // Stage transposed weight chunk: wsT[cl*9+k][co_local].
    // Store lanes sweep co_local fastest -> conflict-free ds stores.
    for (int i = t; i < 144 * 64; i += 256) {
      int col = i & 63;           // co_local
      int cik = i >> 6;           // cl*9 + k
      wsT[cik * 64 + col] = w[(co0 + col) * 1152 + cc * 9 + cik];
    }
    __syncthreads();

    #pragma unroll 1
    for (int cl = 0; cl < 16; ++cl) {
      const float* xrow = &xs[(cc + cl) * 100];
      #pragma unroll
      for (int kh = 0; kh < 3; ++kh) {
        #pragma unroll
        for (int kw = 0; kw < 3; ++kw) {
          const float4 wv = *(const float4*)&wsT[(cl * 9 + kh * 3 + kw) * 64 + cg * 4];
          const float wj[4] = {wv.x, wv.y, wv.z, wv.w};
          const float* xp = xrow + (pr + kh) * 10 + pc + kw;
          const float xv[4] = {xp[0], xp[1], xp[2], xp[3]};
          #pragma unroll
          for (int j = 0; j < 4; ++j)
            #pragma unroll
            for (int i = 0; i < 4; ++i)
              acc[j][i] -= __builtin_fabsf(xv[i] - wj[j]);
        }
      }
    }
  }

  #pragma unroll
  for (int j = 0; j < 4; ++j) {
    int co = co0 + cg * 4 + j;
    float4 o = make_float4(acc[j][0], acc[j][1], acc[j][2], acc[j][3]);
    *(float4*)&out[(((size_t)n * 128 + co) * 64 + (h0 + pr)) * 64 + (w0 + pc)] = o;
  }
}

// ---------------------------------------------------------------- Kernel B --
// Tile: 64 co2 x 64 pixels, K = 128. Weights staged by TDM with +1 dword row
// padding (bank-conflict-free [co][129] layout); activations staged by
// global_load_async_to_lds_b128 into [ci][64pix] (float4 reads).
__global__ __launch_bounds__(256) void adder1x1_kernel(
    const float* __restrict__ mid, const float* __restrict__ w,
    float* __restrict__ out)
{
  __shared__ __align__(16) float ws2[64 * 129];   // 33024 B (TDM row-padded)
  __shared__ __align__(16) float xs2[128 * 64];   // 32768 B

  const int t  = threadIdx.x;
  const int cg = t & 15;
  const int pg = t >> 4;

  int b = blockIdx.x;             // [4 coT][8 n][64 tiles]
  const int coT = b >> 9;  b &= 511;
  const int n   = b >> 6;
  const int tl  = b & 63;
  const int h0  = (tl >> 3) * 8;
  const int w0  = (tl & 7) * 8;
  const int co0 = coT * 64;

  // --- TDM: 2D tile, 64 rows (co2) x 128 floats, row stride 128 floats,
  //     pad_enable: +1 dword every 128 dwords -> LDS rows at stride 129.
  if (t < 32) {  // wave 0 issues the descriptor once
    uint64_t ga  = (uint64_t)(uintptr_t)(w + (size_t)co0 * 128);
    uint32_t lds = (uint32_t)(uintptr_t)&ws2[0];
    v4u g0;
    g0[0] = 1u;                                               // count=1
    g0[1] = lds;                                              // lds_addr
    g0[2] = (uint32_t)ga;                                     // global_addr lo
    g0[3] = ((uint32_t)(ga >> 32) & 0x01FFFFFFu) | (2u << 30); // hi | type=2
    v8i g1;
    g1[0] = (int)((2u << 16)      // data_size = 4B
                | (1u << 20)      // pad_enable
                | (6u << 22));    // pad_interval: every 128 dwords (+1 dword)
    g1[1] = (int)(128u << 16);    // tensor_dim0[15:0] = 128
    g1[2] = (int)(64u << 16);     // tensor_dim1[15:0] = 64
    g1[3] = (int)(128u << 16);    // tile_dim0 = 128
    g1[4] = (int)(64u);           // tile_dim1 = 64, tile_dim2 = 0
    g1[5] = (int)(128u);          // tensor_dim0_stride = 128
    g1[6] = 0;
    g1[7] = 0;
    v4i z4 = {0, 0, 0, 0};
#if defined(__clang_major__) && (__clang_major__ >= 23)
    v8i z8 = {0, 0, 0, 0, 0, 0, 0, 0};
    __builtin_amdgcn_tensor_load_to_lds(g0, g1, z4, z4, z8, 0);
#else
    __builtin_amdgcn_tensor_load_to_lds(g0, g1, z4, z4, 0);
#endif
    __builtin_amdgcn_s_wait_tensorcnt(0);
  }

  // --- Async copy of activation tile: xs2[ci][pix], 16B granules, all
  //     addresses in range (1x1 conv, no halo).
  {
    const uint64_t sbase = (uint64_t)(uintptr_t)(mid + (size_t)n * 128 * 4096);
    const uint32_t lbase = (uint32_t)(uintptr_t)&xs2[0];
    #pragma unroll
    for (int k = 0; k < 8; ++k) {
      int i   = t + k * 256;      // float4-granule index, 0..2047
      int ci  = i >> 4;
      int pq  = i & 15;           // pixel-quad
      int prr = pq >> 1;
      int pcc = (pq & 1) * 4;
      uint32_t voff = (uint32_t)((ci * 4096 + (h0 + prr) * 64 + (w0 + pcc)) * 4);
      uint32_t lad  = lbase + (uint32_t)((ci * 64 + pq * 4) * 4);
      asm volatile("global_load_async_to_lds_b128 %0, %1, %2"
                   :: "v"(lad), "v"(voff), "s"(sbase) : "memory");
    }
    asm volatile("s_wait_asynccnt 0" ::: "memory");
  }
  __syncthreads();

  float acc[4][4];
  #pragma unroll
  for (int j = 0; j < 4; ++j)
    #pragma unroll
    for (int i = 0; i < 4; ++i) acc[j][i] = 0.0f;

  #pragma unroll 2
  for (int ci = 0; ci < 128; ++ci) {
    const float4 xv4 = *(const float4*)&xs2[ci * 64 + pg * 4];
    const float xv[4] = {xv4.x, xv4.y, xv4.z, xv4.w};
    const float* wp = &ws2[(cg * 4) * 129 + ci];
    const float wj[4] = {wp[0], wp[129], wp[258], wp[387]};
    #pragma unroll
    for (int j = 0; j < 4; ++j)
      #pragma unroll
      for (int i = 0; i < 4; ++i)
        acc[j][i] -= __builtin_fabsf(xv[i] - wj[j]);
  }

  #pragma unroll
  for (int j = 0; j < 4; ++j) {
    int co = co0 + cg * 4 + j;
    int hh = h0 + (pg >> 1);
    int wc = w0 + (pg & 1) * 4;
    float4 o = make_float4(acc[j][0], acc[j][1], acc[j][2], acc[j][3]);
    *(float4*)&out[(((size_t)n * 256 + co) * 64 + hh) * 64 + wc] = o;
  }
}

// ------------------------------------------------------------------ launch --
extern "C" void kernel_launch(void* const* d_in, const int* in_sizes, int n_in,
                              void* d_out, int out_size, void* d_ws, size_t ws_size,
                              hipStream_t stream) {
  const float* x    = (const float*)d_in[0];   // [8,128,64,64]
  const float* w_dw = (const float*)d_in[1];   // [128,128,3,3]
  const float* w_pw = (const float*)d_in[2];   // [256,128,1,1]
  float* out = (float*)d_out;                  // [8,256,64,64]
  float* mid = (float*)d_ws;                   // [8,128,64,64] = 16.8 MB

  adder3x3_kernel<<<dim3(2 * 8 * 64), dim3(256), 0, stream>>>(x, w_dw, mid);
  adder1x1_kernel<<<dim3(4 * 8 * 64), dim3(256), 0, stream>>>(mid, w_pw, out);
}